// CompiledConv_61211873902638
// MI455X (gfx1250) — compile-verified
//
#include <hip/hip_runtime.h>
#include <cstdint>

// ---- problem constants (match reference) ----
static constexpr int Bn   = 4;
static constexpr int CIN  = 8;
static constexpr int Hh   = 128;
static constexpr int Ww   = 128;
static constexpr int COUT = 16;
static constexpr int KH   = 5;
static constexpr int KW   = 5;
static constexpr int HO   = Hh - KH + 1;   // 124
static constexpr int WO   = Ww - KW + 1;   // 124

// ---- tiling ----
static constexpr int TX = 64;              // output tile width  (2 px / thread, stride 32)
static constexpr int TY = 8;               // output tile height
static constexpr int TILE_W = TX + KW - 1; // 68
static constexpr int TILE_H = TY + KH - 1; // 12
static constexpr int NTAPS  = KH * KW;     // 25
static constexpr int NW     = NTAPS * CIN * COUT; // 3200 weights

typedef float f32x2 __attribute__((ext_vector_type(2)));

#if defined(__gfx1250__) && __has_builtin(__builtin_amdgcn_tensor_load_to_lds) && __has_builtin(__builtin_amdgcn_s_wait_tensorcnt)
#define USE_TDM 1
typedef unsigned int u32x4 __attribute__((ext_vector_type(4)));
typedef int          i32x4 __attribute__((ext_vector_type(4)));
typedef int          i32x8 __attribute__((ext_vector_type(8)));
#else
#define USE_TDM 0
#endif

__global__ __launch_bounds__(256) void maxplus_conv_gfx1250(
    const float* __restrict__ img,   // [B, CIN, H, W]
    const float* __restrict__ kern,  // [COUT, CIN, KH, KW]
    float* __restrict__ out)         // [B, COUT, HO, WO]
{
  __shared__ float s_tile[CIN * TILE_H * TILE_W]; // 6528 f = 26112 B, layout [c][row][col]
  __shared__ float s_w[NW];                       // 3200 f = 12800 B, layout [tap][o]

  const int tid = threadIdx.x;
  const int bx  = blockIdx.x;   // 0..1
  const int by  = blockIdx.y;   // 0..15
  const int b   = blockIdx.z;   // 0..3
  const int gx0 = bx * TX;
  const int gy0 = by * TY;

  // CDNA5 prefetch (global_prefetch_b8) of the small weight tensor.
  __builtin_prefetch(kern + tid, 0, 1);

  // ---- stage flipped+transposed weights into LDS: s_w[(c*25+i*5+j)*16 + o] ----
  for (int e = tid; e < NW; e += 256) {
    const int o   = e & (COUT - 1);
    const int tap = e >> 4;
    const int c   = tap / NTAPS;
    const int r   = tap % NTAPS;
    const int i   = r / KW;
    const int j   = r % KW;
    s_w[e] = kern[((o * CIN + c) * KH + (KH - 1 - i)) * KW + (KW - 1 - j)];
  }

  // ---- stage image tile (8ch x 12 x 68) into LDS ----
  const float* gtile = img + ((size_t)(b * CIN) * Hh + gy0) * Ww + gx0; // c=0 plane, tile origin

#if USE_TDM
  // One Tensor-Data-Mover descriptor moves the whole 3D tile; OOB elements read as 0
  // (only ever consumed by threads whose outputs are discarded).
  if (tid == 0) {
    const uint64_t ga  = (uint64_t)(uintptr_t)gtile;
    const uint32_t lds = (uint32_t)(uintptr_t)(&s_tile[0]); // low 32b of generic ptr = LDS byte addr
    const uint32_t td0 = (uint32_t)(Ww - gx0);  // remaining cols from tile start (OOB bound)
    const uint32_t td1 = (uint32_t)(Hh - gy0);  // remaining rows
    const uint32_t s0  = (uint32_t)Ww;          // row stride (elements)
    const uint32_t s1  = (uint32_t)(Hh * Ww);   // plane stride (elements)

    // D# group 0: [1:0]=count=1, [63:32]=lds_addr, [120:64]=global_addr, [127:126]=type=2
    u32x4 g0 = { 1u,
                 lds,
                 (uint32_t)ga,
                 (uint32_t)((ga >> 32) & 0x01FFFFFFu) | (2u << 30) };

    // D# group 1 (bit-packed per ISA 8.4)
    i32x8 g1 = {
      (int)(2u << 16),                                  // wg_mask=0, data_size=2 (4B)
      (int)(td0 << 16),                                 // atomic_addr=0 | tensor_dim0[15:0]
      (int)((td0 >> 16) | (td1 << 16)),                 // tensor_dim0[31:16] | tensor_dim1[15:0]
      (int)((td1 >> 16) | ((uint32_t)TILE_W << 16)),    // tensor_dim1[31:16] | tile_dim0=68
      (int)((uint32_t)TILE_H | ((uint32_t)CIN << 16)),  // tile_dim1=12 | tile_dim2=8
      (int)s0,                                          // tensor_dim0_stride[31:0]
      (int)((s1 & 0xFFFFu) << 16),                      // stride0[47:32]=0 | stride1[15:0]
      (int)(s1 >> 16)                                   // tensor_dim1_stride[47:16]
    };

    // D# group 2: tensor_dim2=8, tensor_dim3=0, tensor_dim2_stride=H*W, tile_dim3=0
    i32x4 g2 = { (int)CIN, 0, (int)s1, 0 };
    i32x4 g3 = { 0, 0, 0, 0 };
    // 6th operand on this toolchain (extra int32x8 group) — zero-filled = unused/disabled.
    i32x8 g4 = { 0, 0, 0, 0, 0, 0, 0, 0 };

    __builtin_amdgcn_tensor_load_to_lds(g0, g1, g2, g3, g4, 0);
    __builtin_amdgcn_s_wait_tensorcnt(0);
  }
#else
  for (int t = tid; t < CIN * TILE_H * TILE_W; t += 256) {
    const int c  = t / (TILE_H * TILE_W);
    const int rr = t % (TILE_H * TILE_W);
    const int r  = rr / TILE_W;
    const int x  = rr % TILE_W;
    const int gy = gy0 + r;
    const int gx = gx0 + x;
    float v = 0.0f;
    if (gy < Hh && gx < Ww) v = img[((size_t)(b * CIN + c) * Hh + gy) * Ww + gx];
    s_tile[t] = v;
  }
#endif
  __syncthreads();

  // ---- max-plus accumulation: 2 pixels/thread packed as f32x2 (A = .x, B = .y) ----
  const int ty  = tid >> 5;        // 0..7
  const int tx  = tid & 31;        // 0..31
  const int gy  = gy0 + ty;
  const int gxA = gx0 + tx;        // always < 124
  const int gxB = gxA + 32;        // may be >= 124 at right edge

  f32x2 acc[COUT];
#pragma unroll
  for (int o = 0; o < COUT; ++o) acc[o] = (f32x2){ -3.402823466e38f, -3.402823466e38f };

  for (int c = 0; c < CIN; ++c) {
#pragma unroll
    for (int i = 0; i < KH; ++i) {
      const float* row  = &s_tile[(c * TILE_H + ty + i) * TILE_W + tx];
      const float* wrow = &s_w[(c * NTAPS + i * KW) * COUT];
      // taps (0,1) and (2,3) paired: per o -> 2x v_pk_add_f32 + 2x v_max3_f32
#pragma unroll
      for (int jp = 0; jp < 2; ++jp) {
        const int j0 = jp * 2;
        const f32x2 v0 = { row[j0],     row[32 + j0]     };
        const f32x2 v1 = { row[j0 + 1], row[32 + j0 + 1] };
        const float* w0 = wrow + j0 * COUT;
        const float* w1 = w0 + COUT;
#pragma unroll
        for (int o = 0; o < COUT; ++o) {
          const f32x2 t0 = w0[o] + v0;   // v_pk_add_f32 (scalar splat + packed pixels)
          const f32x2 t1 = w1[o] + v1;   // v_pk_add_f32
          acc[o].x = fmaxf(acc[o].x, fmaxf(t0.x, t1.x));  // v_max3_f32
          acc[o].y = fmaxf(acc[o].y, fmaxf(t0.y, t1.y));  // v_max3_f32
        }
      }
      { // tap j = 4
        const f32x2 v4 = { row[4], row[36] };
        const float* w4 = wrow + 4 * COUT;
#pragma unroll
        for (int o = 0; o < COUT; ++o) {
          const f32x2 t = w4[o] + v4;    // v_pk_add_f32
          acc[o].x = fmaxf(acc[o].x, t.x);
          acc[o].y = fmaxf(acc[o].y, t.y);
        }
      }
    }
  }

  if (gy < HO) {
#pragma unroll
    for (int o = 0; o < COUT; ++o)
      out[((size_t)(b * COUT + o) * HO + gy) * WO + gxA] = acc[o].x;
    if (gxB < WO) {
#pragma unroll
      for (int o = 0; o < COUT; ++o)
        out[((size_t)(b * COUT + o) * HO + gy) * WO + gxB] = acc[o].y;
    }
  }
}

extern "C" void kernel_launch(void* const* d_in, const int* in_sizes, int n_in,
                              void* d_out, int out_size, void* d_ws, size_t ws_size,
                              hipStream_t stream) {
  const float* img  = (const float*)d_in[0];
  const float* kern = (const float*)d_in[1];
  float* out = (float*)d_out;
  (void)in_sizes; (void)n_in; (void)d_ws; (void)ws_size; (void)out_size;

  dim3 grid((WO + TX - 1) / TX, (HO + TY - 1) / TY, Bn); // (2, 16, 4)
  maxplus_conv_gfx1250<<<grid, dim3(256, 1, 1), 0, stream>>>(img, kern, out);
}